// MCT_42167988912850
// MI455X (gfx1250) — compile-verified
//
#include <hip/hip_runtime.h>
#include <hip/hip_bf16.h>
#include <math.h>

// ---------------- constants ----------------
#define NN 1024
#define DD 128

typedef __attribute__((ext_vector_type(2))) float v2f;
typedef __attribute__((ext_vector_type(8))) float v8f;

// ---------------- tiny NN gemm: C[nxn] = A @ B (n=128) ----------------
__global__ void k_gemm_nn_small(const float* __restrict__ A, const float* __restrict__ B,
                                float* __restrict__ C, int n) {
    int idx = blockIdx.x * blockDim.x + threadIdx.x;
    if (idx >= n * n) return;
    int row = idx / n, col = idx % n;
    float s = 0.0f;
    for (int k = 0; k < n; ++k) s += A[row * n + k] * B[k * n + col];
    C[idx] = s;
}

// ---------------- c = w2 @ (w1 @ b0 + b1) + b2  (one block, 128 threads) ----------------
__global__ void k_cvec(const float* __restrict__ b0, const float* __restrict__ b1,
                       const float* __restrict__ b2, const float* __restrict__ w1,
                       const float* __restrict__ w2, float* __restrict__ cvec) {
    __shared__ float t1[DD];
    int t = threadIdx.x;
    float s = b1[t];
    for (int m = 0; m < DD; ++m) s += w1[t * DD + m] * b0[m];
    t1[t] = s;
    __syncthreads();
    float s2 = b2[t];
    for (int k = 0; k < DD; ++k) s2 += w2[t * DD + k] * t1[k];
    cvec[t] = s2;
}

// ---------------- WMMA NT gemm: C[M x Ncols] = A[M x K] @ W[Ncols x K]^T (+bias) ----------
// one 16x16 tile per wave32; K multiple of 4
__global__ void k_gemm_nt_wmma(const float* __restrict__ A, const float* __restrict__ W,
                               const float* __restrict__ bias, float* __restrict__ C,
                               int Ncols, int K) {
    int wave = threadIdx.x >> 5;
    int lane = threadIdx.x & 31;
    int tile = blockIdx.x * (blockDim.x >> 5) + wave;
    int tcols = Ncols >> 4;
    int i0 = (tile / tcols) << 4;
    int j0 = (tile % tcols) << 4;
    int half = lane >> 4, r = lane & 15;

    const float* arow = A + (size_t)(i0 + r) * K + 2 * half;
    const float* brow = W + (size_t)(j0 + r) * K + 2 * half;

    v8f acc = {};
    for (int k = 0; k < K; k += 4) {
        v2f av = *(const v2f*)(arow + k);
        v2f bv = *(const v2f*)(brow + k);
        acc = __builtin_amdgcn_wmma_f32_16x16x4_f32(false, av, false, bv,
                                                    (short)0, acc, false, false);
    }
    float bv = bias ? bias[j0 + r] : 0.0f;
#pragma unroll
    for (int v = 0; v < 8; ++v) {
        int i = i0 + v + 8 * half;
        C[(size_t)i * Ncols + (j0 + r)] = acc[v] + bv;
    }
}

// ---------------- per-row stats: mag, alpha, beta, gamma ----------------
__global__ void k_rowstats(const float* __restrict__ x, const float* __restrict__ a,
                           const float* __restrict__ cvec, float* __restrict__ mag,
                           float* __restrict__ alpha, float* __restrict__ beta,
                           float* __restrict__ gamma) {
    __shared__ float sh[DD];
    int j = blockIdx.x, t = threadIdx.x;
    float xv = x[(size_t)j * DD + t];
    float av = a[(size_t)j * DD + t];
    float d = cvec[t] - xv;
    float vals[4] = {xv * xv, av * av, av * d, d * d};
    float* outs[4] = {mag, alpha, beta, gamma};
    for (int q = 0; q < 4; ++q) {
        sh[t] = vals[q];
        __syncthreads();
        for (int s = DD >> 1; s > 0; s >>= 1) {
            if (t < s) sh[t] += sh[t + s];
            __syncthreads();
        }
        if (t == 0) outs[q][j] = sh[0];
        __syncthreads();
    }
}

// ---------------- WMMA scores = x@x^T fused with affinity epilogue -> Amat ----------------
__global__ void k_affinity_wmma(const float* __restrict__ x, const float* __restrict__ mag,
                                const float* __restrict__ alpha, const float* __restrict__ beta,
                                const float* __restrict__ gamma, float* __restrict__ Amat) {
    int wave = threadIdx.x >> 5;
    int lane = threadIdx.x & 31;
    int tile = blockIdx.x * (blockDim.x >> 5) + wave;
    int tcols = NN >> 4;  // 64
    int i0 = (tile / tcols) << 4;
    int j0 = (tile % tcols) << 4;
    int half = lane >> 4, r = lane & 15;

    const float* arow = x + (size_t)(i0 + r) * DD + 2 * half;
    const float* brow = x + (size_t)(j0 + r) * DD + 2 * half;

    v8f acc = {};
    for (int k = 0; k < DD; k += 4) {
        v2f av = *(const v2f*)(arow + k);
        v2f bv = *(const v2f*)(brow + k);
        acc = __builtin_amdgcn_wmma_f32_16x16x4_f32(false, av, false, bv,
                                                    (short)0, acc, false, false);
    }
    int j = j0 + r;
    float invmag = 1.0f / mag[j];
    float al = alpha[j], be = beta[j], ga = gamma[j];
#pragma unroll
    for (int v = 0; v < 8; ++v) {
        int i = i0 + v + 8 * half;
        float S = acc[v] * invmag;
        float q = S * S * al + 2.0f * S * be + ga;
        float val = __expf(-0.005f * q);  // -0.5/sigma^2 = -0.005
        if (i == j) val = 0.0f;
        Amat[(size_t)i * NN + j] = val;
    }
}

// ---------------- colsum[i] = sum_k A[k,i] ----------------
__global__ void k_colsum(const float* __restrict__ Amat, float* __restrict__ colsum) {
    int i = blockIdx.x * blockDim.x + threadIdx.x;
    float s = 0.0f;
    for (int k = 0; k < NN; ++k) s += Amat[(size_t)k * NN + i];
    colsum[i] = s;
}

// ---------------- p0[j] = A[0,j] / colsum[0] ----------------
__global__ void k_p0(const float* __restrict__ Amat, const float* __restrict__ colsum,
                     float* __restrict__ p0) {
    int j = blockIdx.x * blockDim.x + threadIdx.x;
    p0[j] = Amat[j] / colsum[0];
}

// ---------------- pout[j] = sum_i (pin[i]/colsum[i]) * A[i,j] ----------------
__global__ void k_matvec(const float* __restrict__ Amat, const float* __restrict__ pin,
                         const float* __restrict__ colsum, float* __restrict__ pout) {
    __shared__ float q[DD];
    int j = blockIdx.x * blockDim.x + threadIdx.x;
    float s = 0.0f;
    for (int i0 = 0; i0 < NN; i0 += DD) {
        __syncthreads();
        q[threadIdx.x] = pin[i0 + threadIdx.x] / colsum[i0 + threadIdx.x];
        __syncthreads();
        for (int ii = 0; ii < DD; ++ii) s += q[ii] * Amat[(size_t)(i0 + ii) * NN + j];
    }
    pout[j] = s;
}

// ---------------- softmax over p[1..N-1] -> out[N-1] ----------------
__global__ void k_softmax(const float* __restrict__ p, float* __restrict__ out) {
    __shared__ float sh[NN];
    int t = threadIdx.x;
    float v = (t < NN - 1) ? p[1 + t] : -INFINITY;
    sh[t] = v;
    __syncthreads();
    for (int s = NN >> 1; s > 0; s >>= 1) {
        if (t < s) sh[t] = fmaxf(sh[t], sh[t + s]);
        __syncthreads();
    }
    float mx = sh[0];
    __syncthreads();
    float e = (t < NN - 1) ? __expf(v - mx) : 0.0f;
    sh[t] = e;
    __syncthreads();
    for (int s = NN >> 1; s > 0; s >>= 1) {
        if (t < s) sh[t] += sh[t + s];
        __syncthreads();
    }
    float sum = sh[0];
    if (t < NN - 1) out[t] = e / sum;
}

extern "C" void kernel_launch(void* const* d_in, const int* in_sizes, int n_in,
                              void* d_out, int out_size, void* d_ws, size_t ws_size,
                              hipStream_t stream) {
    const float* f     = (const float*)d_in[0];
    const float* fc1_w = (const float*)d_in[1];
    const float* fc1_b = (const float*)d_in[2];
    const float* e0_w  = (const float*)d_in[3];
    const float* e0_b  = (const float*)d_in[4];
    const float* e1_w  = (const float*)d_in[5];
    const float* e1_b  = (const float*)d_in[6];
    const float* e2_w  = (const float*)d_in[7];
    const float* e2_b  = (const float*)d_in[8];
    float* out = (float*)d_out;

    float* ws = (float*)d_ws;
    size_t o = 0;
    float* x      = ws + o; o += (size_t)NN * DD;
    float* a      = ws + o; o += (size_t)NN * DD;
    float* tmpM   = ws + o; o += (size_t)DD * DD;
    float* Mm     = ws + o; o += (size_t)DD * DD;
    float* cvec   = ws + o; o += DD;
    float* mag    = ws + o; o += NN;
    float* alpha  = ws + o; o += NN;
    float* beta   = ws + o; o += NN;
    float* gamma  = ws + o; o += NN;
    float* colsum = ws + o; o += NN;
    float* p0     = ws + o; o += NN;
    float* p1     = ws + o; o += NN;
    float* Amat   = ws + o; o += (size_t)NN * NN;
    (void)ws_size; (void)in_sizes; (void)n_in; (void)out_size;

    // M = w2 @ w1 @ w0  (tiny 128^3 chains)
    k_gemm_nn_small<<<(DD * DD + 255) / 256, 256, 0, stream>>>(e1_w, e0_w, tmpM, DD);
    k_gemm_nn_small<<<(DD * DD + 255) / 256, 256, 0, stream>>>(e2_w, tmpM, Mm, DD);
    // c = w2 @ (w1 @ b0 + b1) + b2
    k_cvec<<<1, DD, 0, stream>>>(e0_b, e1_b, e2_b, e1_w, e2_w, cvec);

    // x = f @ fc1_w^T + fc1_b      (1024x128, K=128) : 512 tiles / 4 waves = 128 blocks
    k_gemm_nt_wmma<<<128, 128, 0, stream>>>(f, fc1_w, fc1_b, x, DD, DD);
    // a = x @ M^T
    k_gemm_nt_wmma<<<128, 128, 0, stream>>>(x, Mm, nullptr, a, DD, DD);

    // per-row stats
    k_rowstats<<<NN, DD, 0, stream>>>(x, a, cvec, mag, alpha, beta, gamma);

    // A = affinity(x @ x^T)  : 4096 tiles / 4 waves = 1024 blocks
    k_affinity_wmma<<<1024, 128, 0, stream>>>(x, mag, alpha, beta, gamma, Amat);

    // colsum + P0
    k_colsum<<<NN / DD, DD, 0, stream>>>(Amat, colsum);
    k_p0<<<NN / DD, DD, 0, stream>>>(Amat, colsum, p0);

    // 30 random-walk iterations: p <- (p / colsum) @ A
    float* src = p0;
    float* dst = p1;
    for (int it = 0; it < 30; ++it) {
        k_matvec<<<NN / DD, DD, 0, stream>>>(Amat, src, colsum, dst);
        float* t = src; src = dst; dst = t;
    }

    // softmax over P[1:]
    k_softmax<<<1, NN, 0, stream>>>(src, out);
}